// LCN2DLayer_24876450578749
// MI455X (gfx1250) — compile-verified
//
#include <hip/hip_runtime.h>

typedef __attribute__((ext_vector_type(2))) float v2f;
typedef __attribute__((ext_vector_type(8))) float v8f;

#define BATCH 16
#define CIN 8
#define IMG 256
#define NOC 8
#define KW 3
#define TAPS (CIN * KW * KW)      // 72 reduction taps, k = c*9 + i*3 + j
#define WROW (NOC * KW * KW)      // 72 weight floats per (c, pixel) in memory
#define PPB 8                     // pixels (py) per block = wave32s per block

// LDS matrices (per block):
//  pa[pix][m=batch16][k=72]        : A operands, k-contiguous  -> 9216 f (36 KB)
//  wb[pix][n=oc(16, 8 zero)][k=72] : B^T operands, k-contiguous -> 9216 f (36 KB)
#define PA_STRIDE (BATCH * TAPS)          // 1152 per pixel
#define WB_STRIDE (16 * TAPS)             // 1152 per pixel

__global__ __launch_bounds__(256, 2)
void lcn2d_wmma_f32(const float* __restrict__ xin,
                    const float* __restrict__ wgt,
                    const float* __restrict__ bias,
                    float* __restrict__ out)
{
    __shared__ float pa[PPB * PA_STRIDE];   // 36 KB
    __shared__ float wb[PPB * WB_STRIDE];   // 36 KB

    const int tid = threadIdx.x;
    const int px  = blockIdx.y;
    const int py0 = blockIdx.x * PPB;

    // ---- Build wb: stream the BW-dominant weights coalesced (576 contiguous
    // floats per c cover pixels py0..py0+7), scatter-transpose into [n][k].
    for (int c = 0; c < CIN; ++c) {
        const float* src = wgt + ((c * IMG + px) * IMG + py0) * WROW;
        for (int t = tid; t < PPB * WROW; t += 256) {
            int pix = t / WROW;
            int r   = t - pix * WROW;      // oc*9 + tap
            int oc  = r / 9;
            int tap = r - oc * 9;
            wb[pix * WB_STRIDE + oc * TAPS + c * 9 + tap] = src[t];
        }
    }
    // zero-pad columns n = 8..15 (second half of each pixel's 16 x 72 block)
    for (int t = tid; t < PPB * NOC * TAPS; t += 256) {
        int pix = t / (NOC * TAPS);
        int r   = t - pix * (NOC * TAPS);
        wb[pix * WB_STRIDE + NOC * TAPS + r] = 0.0f;
    }

    // ---- Build pa: gather patches (x is L2-resident; symmetric pad == clamp
    // for p=1). One-time decode of k -> (c,i,j) out of the hot loop.
    for (int idx = tid; idx < PPB * PA_STRIDE; idx += 256) {
        int k   = idx % TAPS;
        int m   = (idx / TAPS) & (BATCH - 1);
        int pix = idx / PA_STRIDE;
        int c   = k / 9;
        int r   = k - c * 9;
        int i   = r / 3;
        int j   = r - i * 3;
        int sx = px + i - 1;        sx = sx < 0 ? 0 : (sx > IMG - 1 ? IMG - 1 : sx);
        int sy = py0 + pix - 1 + j; sy = sy < 0 ? 0 : (sy > IMG - 1 ? IMG - 1 : sy);
        pa[idx] = xin[((m * CIN + c) * IMG + sx) * IMG + sy];
    }
    __syncthreads();

    // ---- One wave per pixel: D(16x16) = A(16x72) x B(72x16), 18 x WMMA K=4.
    const int wv   = tid >> 5;          // wave id == local pixel
    const int lane = tid & 31;
    const int py   = py0 + wv;
    const int n    = lane & 15;         // B/D column: oc (valid < 8)
    const int m    = lane & 15;         // A row: batch
    const int kh   = (lane >> 4) << 1;  // per-lane-half K offset {0,2}

    float bv = (n < NOC) ? bias[(px * IMG + py) * NOC + n] : 0.0f;
    v8f acc = { bv, bv, bv, bv, bv, bv, bv, bv };

    // k0 = 4*kk + kh is even -> 8-byte aligned ds_load_b64; kk*16B immediate.
    const float* pa_row = &pa[wv * PA_STRIDE + m * TAPS + kh];
    const float* wb_row = &wb[wv * WB_STRIDE + n * TAPS + kh];

    #pragma unroll
    for (int kk = 0; kk < TAPS / 4; ++kk) {
        v2f a  = *(const v2f*)(pa_row + kk * 4);
        v2f bm = *(const v2f*)(wb_row + kk * 4);
        acc = __builtin_amdgcn_wmma_f32_16x16x4_f32(
            false, a, false, bm, (short)0, acc, false, false);
    }

    // D layout: VGPR r, lanes 0-15 -> M=r; lanes 16-31 -> M=r+8; N = lane&15
    if (n < NOC) {
        const int mb = (lane >> 4) << 3;
        #pragma unroll
        for (int r = 0; r < 8; ++r) {
            int b = r + mb;
            out[((b * NOC + n) * IMG + px) * IMG + py] = acc[r];
        }
    }
}

extern "C" void kernel_launch(void* const* d_in, const int* in_sizes, int n_in,
                              void* d_out, int out_size, void* d_ws, size_t ws_size,
                              hipStream_t stream) {
    const float* x    = (const float*)d_in[0];
    const float* wgt  = (const float*)d_in[1];
    const float* bias = (const float*)d_in[2];
    float* out        = (float*)d_out;

    dim3 grid(IMG / PPB, IMG);   // (32 py-groups, 256 px rows)
    dim3 block(256);             // 8 wave32s = 8 pixels per block
    lcn2d_wmma_f32<<<grid, block, 0, stream>>>(x, wgt, bias, out);
}